// GNNLayer_44770739093651
// MI455X (gfx1250) — compile-verified
//
#include <hip/hip_runtime.h>
#include <hip/hip_bf16.h>

typedef __attribute__((ext_vector_type(2))) float v2f;
typedef __attribute__((ext_vector_type(8))) float v8f;

#define DIM 128   // aggregation feature dim (IN_DIM == OUT_DIM == 128)

// ---------------- degree / normalization ----------------
__global__ void k_init_deg(float* degf, int n) {
    int i = blockIdx.x * blockDim.x + threadIdx.x;
    if (i < n) degf[i] = 1.0f;               // self loop
}

__global__ void k_count_deg(const long long* __restrict__ dst, float* degf, int e) {
    int i = blockIdx.x * blockDim.x + threadIdx.x;
    if (i < e) atomicAdd(&degf[(int)dst[i]], 1.0f);
}

__global__ void k_rsqrt_deg(const float* __restrict__ degf, float* __restrict__ dinv, int n) {
    int i = blockIdx.x * blockDim.x + threadIdx.x;
    if (i < n) {
        float d = degf[i];
        dinv[i] = (d > 0.0f) ? rsqrtf(d) : 0.0f;
    }
}

// ---------------- aggregation: out = A_norm @ f  (128-dim) ----------------
// self-loop term (also initializes out): out[i] = f[i] * dinv[i]^2
__global__ void k_self_init(const float* __restrict__ f, const float* __restrict__ dinv,
                            float* __restrict__ out, int n) {
    long long t = blockIdx.x * (long long)blockDim.x + threadIdx.x;
    int i = (int)(t >> 5);
    int c = ((int)t & 31) << 2;
    if (i >= n) return;
    float w = dinv[i]; w *= w;
    const float4 v = *(const float4*)(f + (size_t)i * DIM + c);
    float4 o = { v.x * w, v.y * w, v.z * w, v.w * w };
    *(float4*)(out + (size_t)i * DIM + c) = o;
}

// edge term: out[dst] += f[src] * dinv[src]*dinv[dst]   (32 lanes x float4 per edge)
__global__ void k_agg_edges(const float* __restrict__ f, const long long* __restrict__ src,
                            const long long* __restrict__ dst, const float* __restrict__ dinv,
                            float* __restrict__ out, int e) {
    long long t = blockIdx.x * (long long)blockDim.x + threadIdx.x;
    int idx = (int)(t >> 5);
    int c = ((int)t & 31) << 2;
    if (idx >= e) return;
    int s = (int)src[idx];
    int d = (int)dst[idx];
    float nrm = dinv[s] * dinv[d];
    const float4 v = *(const float4*)(f + (size_t)s * DIM + c);
    float* o = out + (size_t)d * DIM + c;
    atomicAdd(o + 0, v.x * nrm);
    atomicAdd(o + 1, v.y * nrm);
    atomicAdd(o + 2, v.z * nrm);
    atomicAdd(o + 3, v.w * nrm);
}

// ---------------- WMMA f32 GEMM: C[M,N] = A[M,K] @ W[K,N] (+bias, relu) ----------------
#define BM 128
#define BN 128
#define BK 16
#define LDT (BK + 4)   // 20 dwords = 80B row stride: 8B aligned, conflict-free frag reads

__global__ __launch_bounds__(256)
void k_gemm_wmma(const float* __restrict__ A, const float* __restrict__ W,
                 float* __restrict__ C, const float* __restrict__ bias,
                 int M, int K, int N, int relu) {
    __shared__ float As[BM][LDT];   // K-contiguous per row -> ds_load_b64 fragments
    __shared__ float Bs[BN][LDT];

    const int t      = threadIdx.x;
    const int lane   = t & 31;
    const int wave   = t >> 5;
    const int half   = lane >> 4;      // 0: K pair {0,1}; 1: K pair {2,3}
    const int lm     = lane & 15;
    const int wave_m = (wave & 3) * 32;   // 4 waves along M
    const int wave_n = (wave >> 2) * 64;  // 2 waves along N
    const int m0     = blockIdx.x * BM;
    const int n0     = blockIdx.y * BN;

    v8f acc[2][4] = {};

    for (int k0 = 0; k0 < K; k0 += BK) {
        // --- load A tile: rows stay K-contiguous, single b128 LDS store ---
        #pragma unroll
        for (int p = 0; p < 2; ++p) {
            int idx = t + p * 256;         // 0..511 float4s
            int ml  = idx >> 2;            // 0..127
            int kl  = (idx & 3) << 2;      // 0,4,8,12
            float4 v = { 0.f, 0.f, 0.f, 0.f };
            int gm = m0 + ml;
            if (gm < M) v = *(const float4*)(A + (size_t)gm * K + k0 + kl);
            *(float4*)&As[ml][kl] = v;
        }
        // --- load B tile: W[k][n] row-major -> transpose into Bs[n][k] ---
        #pragma unroll
        for (int p = 0; p < 2; ++p) {
            int idx = t + p * 256;
            int kl  = idx >> 5;            // 0..15
            int nl  = (idx & 31) << 2;     // 0..124
            float4 v = *(const float4*)(W + (size_t)(k0 + kl) * N + n0 + nl);
            Bs[nl + 0][kl] = v.x;
            Bs[nl + 1][kl] = v.y;
            Bs[nl + 2][kl] = v.z;
            Bs[nl + 3][kl] = v.w;
        }
        // prefetch next W tile (always in-bounds when taken)
        if (k0 + BK < K)
            __builtin_prefetch(W + (size_t)(k0 + BK + (t >> 4)) * N + n0 + ((t & 15) << 3), 0, 1);
        __syncthreads();

        // --- 32 WMMAs per K-tile; fragments are aligned v2f -> ds_load_b64 ---
        #pragma unroll
        for (int kk = 0; kk < BK; kk += 4) {
            const int kb = kk + half * 2;
            v2f a[2], b[4];
            #pragma unroll
            for (int mi = 0; mi < 2; ++mi)
                a[mi] = *(const v2f*)&As[wave_m + mi * 16 + lm][kb];
            #pragma unroll
            for (int ni = 0; ni < 4; ++ni)
                b[ni] = *(const v2f*)&Bs[wave_n + ni * 16 + lm][kb];
            #pragma unroll
            for (int mi = 0; mi < 2; ++mi)
                #pragma unroll
                for (int ni = 0; ni < 4; ++ni)
                    acc[mi][ni] = __builtin_amdgcn_wmma_f32_16x16x4_f32(
                        false, a[mi], false, b[ni], (short)0, acc[mi][ni], false, false);
        }
        __syncthreads();
    }

    // --- epilogue: C layout lane = N%16 + 16*(M-half), vgpr r = M%8 ---
    #pragma unroll
    for (int mi = 0; mi < 2; ++mi) {
        #pragma unroll
        for (int ni = 0; ni < 4; ++ni) {
            int nn = n0 + wave_n + ni * 16 + lm;
            float bv = bias ? bias[nn] : 0.0f;
            #pragma unroll
            for (int r = 0; r < 8; ++r) {
                int mm = m0 + wave_m + mi * 16 + half * 8 + r;
                if (mm < M) {
                    float v = acc[mi][ni][r] + bv;
                    if (relu) v = fmaxf(v, 0.0f);
                    C[(size_t)mm * N + nn] = v;
                }
            }
        }
    }
}

// ---------------- fused bias + layernorm (one wave32 per 128-elem row, in place) ----------------
__global__ void k_bias_ln(float* __restrict__ out, const float* __restrict__ b,
                          const float* __restrict__ gamma, const float* __restrict__ beta, int n) {
    long long t = blockIdx.x * (long long)blockDim.x + threadIdx.x;
    int row  = (int)(t >> 5);
    int lane = (int)t & 31;
    if (row >= n) return;
    float* r = out + (size_t)row * DIM;
    int c = lane << 2;
    float4 v = *(float4*)(r + c);
    const float4 bb = *(const float4*)(b + c);
    v.x += bb.x; v.y += bb.y; v.z += bb.z; v.w += bb.w;

    float s = v.x + v.y + v.z + v.w;
    #pragma unroll
    for (int m = 16; m >= 1; m >>= 1) s += __shfl_xor(s, m, 32);
    float mu = s * (1.0f / DIM);

    float dx = v.x - mu, dy = v.y - mu, dz = v.z - mu, dw = v.w - mu;
    float q = dx * dx + dy * dy + dz * dz + dw * dw;
    #pragma unroll
    for (int m = 16; m >= 1; m >>= 1) q += __shfl_xor(q, m, 32);
    float rs = rsqrtf(q * (1.0f / DIM) + 1e-5f);

    const float4 g  = *(const float4*)(gamma + c);
    const float4 be = *(const float4*)(beta + c);
    float4 o = { dx * rs * g.x + be.x, dy * rs * g.y + be.y,
                 dz * rs * g.z + be.z, dw * rs * g.w + be.w };
    *(float4*)(r + c) = o;
}

// ---------------- launcher ----------------
extern "C" void kernel_launch(void* const* d_in, const int* in_sizes, int n_in,
                              void* d_out, int out_size, void* d_ws, size_t ws_size,
                              hipStream_t stream) {
    (void)n_in; (void)out_size; (void)ws_size;
    const float*     x     = (const float*)d_in[0];
    const long long* ei    = (const long long*)d_in[1];   // int64 [2, E]
    const float*     W1    = (const float*)d_in[2];
    const float*     b1    = (const float*)d_in[3];
    const float*     W2    = (const float*)d_in[4];
    const float*     b2    = (const float*)d_in[5];
    const float*     gamma = (const float*)d_in[6];
    const float*     beta  = (const float*)d_in[7];
    float*           out   = (float*)d_out;

    const int n = in_sizes[0] / 128;
    const int e = in_sizes[1] / 2;
    const int IN = 128, HID = 256, OUT = 128;
    const long long* src = ei;
    const long long* dst = ei + e;

    char*  ws   = (char*)d_ws;
    float* degf = (float*)ws;                                   // n
    float* dinv = degf + n;                                     // n
    size_t off  = ((size_t)2 * n * sizeof(float) + 255) & ~(size_t)255;
    float* bufA = (float*)(ws + off);                           // n*128 (agg1, later z)
    float* bufH = bufA + (size_t)n * 128;                       // n*256 (h1)

    // symmetric-normalization degrees (in-degree + self-loop)
    k_init_deg <<<(n + 255) / 256, 256, 0, stream>>>(degf, n);
    k_count_deg<<<(e + 255) / 256, 256, 0, stream>>>(dst, degf, e);
    k_rsqrt_deg<<<(n + 255) / 256, 256, 0, stream>>>(degf, dinv, n);

    const int gn = (int)(((long long)n * 32 + 255) / 256);
    const int ge = (int)(((long long)e * 32 + 255) / 256);

    // layer 1: agg1 = A_norm @ x ; h1 = relu(agg1 @ W1 + b1)
    k_self_init<<<gn, 256, 0, stream>>>(x, dinv, bufA, n);
    k_agg_edges<<<ge, 256, 0, stream>>>(x, src, dst, dinv, bufA, e);
    dim3 g1((n + BM - 1) / BM, HID / BN);
    k_gemm_wmma<<<g1, 256, 0, stream>>>(bufA, W1, bufH, b1, n, IN, HID, 1);

    // layer 2: z = h1 @ W2 ; out = layernorm(A_norm @ z + b2)
    dim3 g2((n + BM - 1) / BM, OUT / BN);
    k_gemm_wmma<<<g2, 256, 0, stream>>>(bufH, W2, bufA, nullptr, n, HID, OUT, 0);
    k_self_init<<<gn, 256, 0, stream>>>(bufA, dinv, out, n);
    k_agg_edges<<<ge, 256, 0, stream>>>(bufA, src, dst, dinv, out, e);
    k_bias_ln  <<<gn, 256, 0, stream>>>(out, b2, gamma, beta, n);
}